// TopologyGNN_51144470561341
// MI455X (gfx1250) — compile-verified
//
#include <hip/hip_runtime.h>
#include <hip/hip_bf16.h>

typedef float v2f __attribute__((ext_vector_type(2)));
typedef float v8f __attribute__((ext_vector_type(8)));

// ---------------------------------------------------------------------------
// Fragment group loader for one K-step (4 columns of A, 4 16-wide B tiles).
// BT=true : B stored (N,K) row-major -> op(B)=B^T  (for X @ W.T), float2 loads
// BT=false: B stored (K,N) row-major -> op(B)=B    (for X @ W), scalar loads
// ---------------------------------------------------------------------------
template<bool BT>
__device__ __forceinline__ void load_frags(const float* __restrict__ Ab,
                                           const float* __restrict__ Bb,
                                           int row, int c0, int kb, int K, int N,
                                           v2f& a, v2f& b0, v2f& b1, v2f& b2, v2f& b3)
{
    a = *(const v2f*)(Ab + (size_t)row * K + kb);           // 8B aligned (K even)
    if constexpr (BT) {
        b0 = *(const v2f*)(Bb + (size_t)(c0     ) * K + kb);
        b1 = *(const v2f*)(Bb + (size_t)(c0 + 16) * K + kb);
        b2 = *(const v2f*)(Bb + (size_t)(c0 + 32) * K + kb);
        b3 = *(const v2f*)(Bb + (size_t)(c0 + 48) * K + kb);
    } else {
        const float* r0 = Bb + (size_t)kb * N;
        const float* r1 = r0 + N;
        b0.x = r0[c0     ];  b0.y = r1[c0     ];
        b1.x = r0[c0 + 16];  b1.y = r1[c0 + 16];
        b2.x = r0[c0 + 32];  b2.y = r1[c0 + 32];
        b3.x = r0[c0 + 48];  b3.y = r1[c0 + 48];
    }
}

// ---------------------------------------------------------------------------
// fp32 WMMA GEMM: C[M,N] = A[M,K] * op(B) (+ bias[N]), optional batching.
// One wave computes a 16x64 C slab (4 accumulators share each A fragment).
// K loop is software-pipelined one group deep so fragment loads for step
// kk+4 are in flight while the four V_WMMA_F32_16X16X4_F32 of step kk run.
// ---------------------------------------------------------------------------
template<bool BT>
__global__ void wmma_gemm_f32(const float* __restrict__ A,
                              const float* __restrict__ B,
                              float* __restrict__ C,
                              const float* __restrict__ bias,
                              int M, int N, int K,
                              long long sA, long long sB, long long sC)
{
    const float* Ab = A + sA * blockIdx.z;
    const float* Bb = B + sB * blockIdx.z;
    float*       Cb = C + sC * blockIdx.z;

    const int lane = threadIdx.x & 31;
    const int i0   = (blockIdx.y * blockDim.y + threadIdx.y) * 16; // 16 rows
    const int j0   = blockIdx.x * 64;                              // 64 cols
    if (i0 >= M || j0 >= N) return;

    const int l   = lane & 15;
    const int hi  = lane >> 4;          // 0: K pair (kk,kk+1)   1: (kk+2,kk+3)
    const int row = i0 + l;             // A row for this lane
    const int c0  = j0 + l;             // base B/C col for this lane

    v8f acc0 = {}, acc1 = {}, acc2 = {}, acc3 = {};

    v2f a, b0, b1, b2, b3;
    load_frags<BT>(Ab, Bb, row, c0, (hi << 1), K, N, a, b0, b1, b2, b3);

    #pragma unroll 2
    for (int kk = 0; kk < K - 4; kk += 4) {
        v2f an, n0, n1, n2, n3;
        load_frags<BT>(Ab, Bb, row, c0, kk + 4 + (hi << 1), K, N,
                       an, n0, n1, n2, n3);
        acc0 = __builtin_amdgcn_wmma_f32_16x16x4_f32(false, a, false, b0, (short)0, acc0, false, false);
        acc1 = __builtin_amdgcn_wmma_f32_16x16x4_f32(false, a, false, b1, (short)0, acc1, false, false);
        acc2 = __builtin_amdgcn_wmma_f32_16x16x4_f32(false, a, false, b2, (short)0, acc2, false, false);
        acc3 = __builtin_amdgcn_wmma_f32_16x16x4_f32(false, a, false, b3, (short)0, acc3, false, false);
        a = an; b0 = n0; b1 = n1; b2 = n2; b3 = n3;
    }
    acc0 = __builtin_amdgcn_wmma_f32_16x16x4_f32(false, a, false, b0, (short)0, acc0, false, false);
    acc1 = __builtin_amdgcn_wmma_f32_16x16x4_f32(false, a, false, b1, (short)0, acc1, false, false);
    acc2 = __builtin_amdgcn_wmma_f32_16x16x4_f32(false, a, false, b2, (short)0, acc2, false, false);
    acc3 = __builtin_amdgcn_wmma_f32_16x16x4_f32(false, a, false, b3, (short)0, acc3, false, false);

    #pragma unroll
    for (int r = 0; r < 8; ++r) {
        const int cr = i0 + r + hi * 8;
        float* crow = Cb + (size_t)cr * N;
        float v0 = acc0[r], v1 = acc1[r], v2 = acc2[r], v3 = acc3[r];
        if (bias) {
            v0 += bias[c0     ];
            v1 += bias[c0 + 16];
            v2 += bias[c0 + 32];
            v3 += bias[c0 + 48];
        }
        crow[c0     ] = v0;
        crow[c0 + 16] = v1;
        crow[c0 + 32] = v2;
        crow[c0 + 48] = v3;
    }
}

// ---------------------------------------------------------------------------
// GRU cell update: h = (1-z)*n + z*h  (elementwise over BN x D)
// gi/gh are (BN, 3D) with layout [r | z | n].
// ---------------------------------------------------------------------------
__global__ void gru_update(const float* __restrict__ gi,
                           const float* __restrict__ gh,
                           float* __restrict__ h, int D)
{
    const size_t idx  = (size_t)blockIdx.x * blockDim.x + threadIdx.x;
    const size_t i    = idx / D, d = idx % D;
    const size_t base = i * (size_t)(3 * D);
    const float ir = gi[base + d], iz = gi[base + D + d], in = gi[base + 2 * D + d];
    const float hr = gh[base + d], hz = gh[base + D + d], hn = gh[base + 2 * D + d];
    const float r = 1.f / (1.f + expf(-(ir + hr)));
    const float z = 1.f / (1.f + expf(-(iz + hz)));
    const float n = tanhf(in + r * hn);
    h[idx] = (1.f - z) * n + z * h[idx];
}

// ---------------------------------------------------------------------------
// LayerNorm over D=256, in place. One 256-thread block per row.
// var is population variance (matches jnp.var).
// ---------------------------------------------------------------------------
__global__ void layernorm256(float* __restrict__ x,
                             const float* __restrict__ g,
                             const float* __restrict__ b, int D)
{
    __shared__ float sh1[8], sh2[8];
    __shared__ float mu_s, rs_s;
    const int row = blockIdx.x, t = threadIdx.x;
    const float v = x[(size_t)row * D + t];
    float s1 = v, s2 = v * v;
    #pragma unroll
    for (int off = 16; off > 0; off >>= 1) {
        s1 += __shfl_xor(s1, off, 32);
        s2 += __shfl_xor(s2, off, 32);
    }
    if ((t & 31) == 0) { sh1[t >> 5] = s1; sh2[t >> 5] = s2; }
    __syncthreads();
    if (t == 0) {
        float a1 = 0.f, a2 = 0.f;
        #pragma unroll
        for (int i = 0; i < 8; ++i) { a1 += sh1[i]; a2 += sh2[i]; }
        const float mu  = a1 / (float)D;
        const float var = a2 / (float)D - mu * mu;
        mu_s = mu;
        rs_s = rsqrtf(var + 1e-5f);
    }
    __syncthreads();
    x[(size_t)row * D + t] = (v - mu_s) * rs_s * g[t] + b[t];
}

// ---------------------------------------------------------------------------
// Edge predictor, LDS-tiled:
//   logits[b,n,m] = sum_d relu(a[b,n,d] + bb[b,m,d] + eb1[d]) * w2[d] + eb2
//   adj = sigmoid(logits)
// One 256-thread block computes a 32x32 (n,m) tile. Both 32x256 row tiles
// are staged in LDS (stride 257 -> conflict-free; ~67 KB of the WGP's 320 KB).
// Each thread register-blocks 4 outputs (fixed n, 4 consecutive m), so LDS
// reads amortize to 1.5 per fma. ReLU inside the contraction forbids WMMA.
// ---------------------------------------------------------------------------
__global__ void edge_predict(const float* __restrict__ amat,
                             const float* __restrict__ bmat,
                             const float* __restrict__ eb1,
                             const float* __restrict__ w2,
                             const float* __restrict__ eb2,
                             float* __restrict__ logits,
                             float* __restrict__ adj)
{
    constexpr int N = 256, D = 256, T = 32, LDSW = 257;
    __shared__ float sa[T][LDSW];
    __shared__ float sb[T][LDSW];
    __shared__ float sw[D];

    const int bi  = blockIdx.x;          // b*64 + (n-tile)*8 + (m-tile)
    const int b   = bi >> 6;
    const int n0  = ((bi >> 3) & 7) * T;
    const int m0  = (bi & 7) * T;
    const int t   = threadIdx.x;

    // stage tiles: 32 rows x 256 cols each, coalesced
    for (int idx = t; idx < T * D; idx += 256) {
        const int i = idx >> 8, d = idx & 255;
        sa[i][d] = amat[((size_t)b * N + n0 + i) * D + d] + eb1[d];
        sb[i][d] = bmat[((size_t)b * N + m0 + i) * D + d];
    }
    sw[t] = w2[t];
    __syncthreads();

    const int ni  = t >> 3;        // 0..31
    const int mi0 = (t & 7) * 4;   // 0,4,...,28
    float a0 = 0.f, a1 = 0.f, a2 = 0.f, a3 = 0.f;
    #pragma unroll 4
    for (int d = 0; d < D; ++d) {
        const float av = sa[ni][d];     // broadcast across 8 lanes
        const float wv = sw[d];
        a0 += fmaxf(av + sb[mi0    ][d], 0.f) * wv;
        a1 += fmaxf(av + sb[mi0 + 1][d], 0.f) * wv;
        a2 += fmaxf(av + sb[mi0 + 2][d], 0.f) * wv;
        a3 += fmaxf(av + sb[mi0 + 3][d], 0.f) * wv;
    }

    const float c2 = eb2[0];
    const size_t base = ((size_t)b * N + n0 + ni) * N + m0 + mi0;
    float lg[4] = { a0 + c2, a1 + c2, a2 + c2, a3 + c2 };
    #pragma unroll
    for (int j = 0; j < 4; ++j) {
        logits[base + j] = lg[j];
        adj[base + j]    = 1.f / (1.f + expf(-lg[j]));
    }
}

extern "C" void kernel_launch(void* const* d_in, const int* in_sizes, int n_in,
                              void* d_out, int out_size, void* d_ws, size_t ws_size,
                              hipStream_t stream)
{
    (void)in_sizes; (void)n_in; (void)out_size; (void)ws_size;
    const float* x        = (const float*)d_in[0];
    const float* init_adj = (const float*)d_in[1];
    const float* gcn_w    = (const float*)d_in[2];
    const float* gcn_b    = (const float*)d_in[3];
    const float* w_ih     = (const float*)d_in[4];
    const float* w_hh     = (const float*)d_in[5];
    const float* b_ih     = (const float*)d_in[6];
    const float* b_hh     = (const float*)d_in[7];
    const float* ln_g     = (const float*)d_in[8];
    const float* ln_b     = (const float*)d_in[9];
    const float* ep_w1    = (const float*)d_in[10];
    const float* ep_b1    = (const float*)d_in[11];
    const float* ep_w2    = (const float*)d_in[12];
    const float* ep_b2    = (const float*)d_in[13];

    const int B = 8, N = 256, D = 256, BN = B * N;          // BN = 2048
    const size_t NN = (size_t)N * N;                        // 65536

    float* ws = (float*)d_ws;
    float* gi = ws;                                         // BN * 3D
    float* gh = gi + (size_t)BN * 3 * D;                    // BN * 3D
    float* h  = gh + (size_t)BN * 3 * D;                    // BN * D
    float* tt = h  + (size_t)BN * D;                        // BN * D   (h @ gcn_w)
    float* f  = tt + (size_t)BN * D;                        // BN * D   (gcn -> LN in place)
    float* av = f  + (size_t)BN * D;                        // BN * D   (f @ w1a)
    float* bv = av + (size_t)BN * D;                        // BN * D   (f @ w1b)

    float* out     = (float*)d_out;
    float* adj     = out;                                   // B*N*N (live adjacency)
    float* logits0 = out + (size_t)B * NN;                  // 6 * B*N*N

    hipMemcpyAsync(adj, init_adj, sizeof(float) * B * NN,
                   hipMemcpyDeviceToDevice, stream);
    hipMemsetAsync(h, 0, sizeof(float) * BN * D, stream);

    const dim3 blk(32, 4);                                  // 4 waves; 16x64 slab each
    const dim3 gBig(3 * D / 64, BN / 16 / 4, 1);            // 12 x 32  (M=2048,N=768)
    const dim3 gSqr(D / 64,     BN / 16 / 4, 1);            //  4 x 32  (M=2048,N=256)
    const dim3 gAdj(D / 64,     N  / 16 / 4, B);            //  4 x 4 x 8 (batched)
    const int  gEdge = B * (N / 32) * (N / 32);             // 512 blocks

    // gi = x @ w_ih^T + b_ih   (once)
    wmma_gemm_f32<true><<<gBig, blk, 0, stream>>>(x, w_ih, gi, b_ih,
                                                  BN, 3 * D, D, 0, 0, 0);
    for (int L = 0; L < 6; ++L) {
        // gh = h @ w_hh^T + b_hh
        wmma_gemm_f32<true><<<gBig, blk, 0, stream>>>(h, w_hh, gh, b_hh,
                                                      BN, 3 * D, D, 0, 0, 0);
        // GRU: h = (1-z)*n + z*h
        gru_update<<<BN * D / 256, 256, 0, stream>>>(gi, gh, h, D);
        // tt = h @ gcn_w
        wmma_gemm_f32<false><<<gSqr, blk, 0, stream>>>(h, gcn_w, tt, nullptr,
                                                       BN, D, D, 0, 0, 0);
        // f[b] = adj[b] @ tt[b] + gcn_b   (batched over B)
        wmma_gemm_f32<false><<<gAdj, blk, 0, stream>>>(adj, tt, f, gcn_b,
                                                       N, D, N,
                                                       (long long)NN,
                                                       (long long)N * D,
                                                       (long long)N * D);
        // f = LayerNorm(f)
        layernorm256<<<BN, 256, 0, stream>>>(f, ln_g, ln_b, D);
        // av = f @ w1a ; bv = f @ w1b   (w1a = ep_w1[:D], w1b = ep_w1[D:])
        wmma_gemm_f32<false><<<gSqr, blk, 0, stream>>>(f, ep_w1, av, nullptr,
                                                       BN, D, D, 0, 0, 0);
        wmma_gemm_f32<false><<<gSqr, blk, 0, stream>>>(f, ep_w1 + (size_t)D * D,
                                                       bv, nullptr,
                                                       BN, D, D, 0, 0, 0);
        // logits + adj = sigmoid(logits)
        edge_predict<<<gEdge, 256, 0, stream>>>(av, bv, ep_b1, ep_w2, ep_b2,
                                                logits0 + (size_t)L * B * NN,
                                                adj);
    }
}